// FMFM_WTA_Network_42262478192720
// MI455X (gfx1250) — compile-verified
//
#include <hip/hip_runtime.h>
#include <hip/hip_bf16.h>

typedef __attribute__((ext_vector_type(2))) float v2f;
typedef __attribute__((ext_vector_type(8))) float v8f;

#define T_STEPS 1024
#define B_SIZE  1024
#define N_NEUR  64
#define BETA    0.9f
#define THRESH  1.0f

#define TB ((size_t)T_STEPS * B_SIZE)

// ---------------------------------------------------------------------------
// Kernel 1: bank LIF simulation via V_WMMA_F32_16X16X4_F32.
// Block = 128 threads (4 waves). Wave w owns neuron tile n in [16w, 16w+16).
// Block x owns batch tile b in [16x, 16x+16).
// Membrane update fused into the WMMA:  mem' = A(x_t)*B(W^T) + (beta*mem - reset)
// Spike bits extracted with ballots; each wave writes 16 contiguous u16 pieces
// into its own mask plane:  plane[w][t*B + b]  (bit j = spk(b, 16w+j)).
// A-loads are software-pipelined one timestep ahead.
// ---------------------------------------------------------------------------
__global__ __launch_bounds__(128)
void k_bank(const float* __restrict__ spike, const float* __restrict__ W,
            unsigned short* __restrict__ maskp)
{
    const int lane = threadIdx.x & 31;
    const int wave = threadIdx.x >> 5;       // neuron tile 0..3
    const int b0   = blockIdx.x * 16;        // batch tile base

    unsigned short* plane = maskp + (size_t)wave * TB;

    // B matrix (4x16, K x N): K=0 row = w1[n], K=1 row = w2[n], K=2,3 = 0.
    v2f bm;
    if (lane < 16) {
        const int n = wave * 16 + lane;
        bm.x = W[n * 2 + 0];
        bm.y = W[n * 2 + 1];
    } else {
        bm.x = 0.0f; bm.y = 0.0f;
    }

    // A matrix loader (16x4, M x K): row M = b0+M, K=0 -> x0, K=1 -> x1, K=2,3 -> 0
    auto loadA = [&](int t) -> v2f {
        v2f a;
        if (lane < 16) {
            const float* xp = spike + ((size_t)t * B_SIZE + b0 + lane) * 2;
            a.x = xp[0]; a.y = xp[1];
        } else {
            a.x = 0.0f; a.y = 0.0f;
        }
        return a;
    };

    v8f mem = {};          // C/D layout: (M = i + 8*(lane>=16), N = lane%16)
    v2f a = loadA(0);
    const unsigned v = (unsigned)(lane & 7);   // which ballot holds this lane's row

    for (int t = 0; t < T_STEPS; ++t) {
        // issue next A-load early to hide latency behind leak + WMMA
        const int tn = (t + 1 < T_STEPS) ? (t + 1) : (T_STEPS - 1);
        v2f an = loadA(tn);

        // C = beta*mem - reset(prev mem)
        v8f c;
        #pragma unroll
        for (int i = 0; i < 8; ++i) {
            const float m = mem[i];
            c[i] = BETA * m - ((m > THRESH) ? THRESH : 0.0f);
        }

        // mem' = x_t @ W^T + C    (exact f32, K zero-padded)
        mem = __builtin_amdgcn_wmma_f32_16x16x4_f32(
            /*neg_a=*/false, a, /*neg_b=*/false, bm,
            /*c_mod=*/(short)0, c, /*reuse_a=*/false, /*reuse_b=*/false);

        // ballot(vgpr i): bits[0:16) = (M=i, N=0..15), bits[16:32) = (M=i+8, N=0..15)
        unsigned bal[8];
        #pragma unroll
        for (int i = 0; i < 8; ++i)
            bal[i] = (unsigned)__ballot(mem[i] > THRESH);

        // branch-free select: word = bal[lane & 7]
        unsigned word = 0u;
        #pragma unroll
        for (int i = 0; i < 8; ++i)
            word |= (v == (unsigned)i) ? bal[i] : 0u;

        if (lane < 16) {
            const unsigned piece = (lane < 8) ? (word & 0xFFFFu) : (word >> 16);
            plane[(size_t)t * B_SIZE + b0 + lane] = (unsigned short)piece;
        }

        a = an;
    }
}

__device__ __forceinline__ unsigned long long load_mask(
    const unsigned short* __restrict__ maskp, size_t tb)
{
    const unsigned long long p0 = maskp[tb];
    const unsigned long long p1 = maskp[tb + TB];
    const unsigned long long p2 = maskp[tb + 2 * TB];
    const unsigned long long p3 = maskp[tb + 3 * TB];
    return p0 | (p1 << 16) | (p2 << 32) | (p3 << 48);
}

// ---------------------------------------------------------------------------
// Kernel 2: active[t] = OR over b of mask[t][b]; also zero fire[t].
// ---------------------------------------------------------------------------
__global__ __launch_bounds__(256)
void k_active(const unsigned short* __restrict__ maskp,
              unsigned long long* __restrict__ active,
              unsigned* __restrict__ fire)
{
    __shared__ unsigned long long s[256];
    const int t = blockIdx.x;
    unsigned long long acc = 0ull;
    for (int i = threadIdx.x; i < B_SIZE; i += 256)
        acc |= load_mask(maskp, (size_t)t * B_SIZE + i);
    s[threadIdx.x] = acc;
    __syncthreads();
    for (int off = 128; off > 0; off >>= 1) {
        if (threadIdx.x < off) s[threadIdx.x] |= s[threadIdx.x + off];
        __syncthreads();
    }
    if (threadIdx.x == 0) { active[t] = s[0]; fire[t] = 0u; }
}

// ---------------------------------------------------------------------------
// Kernel 3: un-inhibited readout LIF per batch; mark fire[t] when any b fires.
// One block of 1024 threads (thread = b).
// ---------------------------------------------------------------------------
__global__ __launch_bounds__(1024)
void k_ro_fire(const unsigned short* __restrict__ maskp,
               unsigned* __restrict__ fire)
{
    const int b = threadIdx.x;
    const int lane = b & 31;
    float m = 0.0f;
    for (int t = 0; t < T_STEPS; ++t) {
        const unsigned long long msk = load_mask(maskp, (size_t)t * B_SIZE + b);
        const float cur   = (float)__popcll(msk);
        const float reset = (m > THRESH) ? THRESH : 0.0f;
        m = BETA * m + cur - reset;
        const unsigned long long any = __ballot(m > THRESH);
        if (any && lane == 0) atomicOr(&fire[t], 1u);
    }
}

// ---------------------------------------------------------------------------
// Kernel 4: scalars. t_f = first t with fire; winner = min-delay active neuron
// at the first t <= t_f with any activity.
// ---------------------------------------------------------------------------
__global__ void k_scalars(const unsigned long long* __restrict__ active,
                          const unsigned* __restrict__ fire,
                          const int* __restrict__ delays,
                          int* __restrict__ scal,
                          float* __restrict__ out_tail)
{
    if (threadIdx.x != 0 || blockIdx.x != 0) return;
    int tf = -1;
    for (int t = 0; t < T_STEPS; ++t)
        if (fire[t]) { tf = t; break; }
    int winner = -1;
    const int lim = (tf < 0) ? (T_STEPS - 1) : tf;
    for (int t = 0; t <= lim; ++t) {
        const unsigned long long a = active[t];
        if (a) {
            int best = -1, bestd = 0x7fffffff;
            for (int n = 0; n < N_NEUR; ++n)
                if ((a >> n) & 1ull) {
                    const int d = delays[n];
                    if (d < bestd) { bestd = d; best = n; }
                }
            winner = best;
            break;
        }
    }
    scal[0] = tf;
    scal[1] = winner;
    out_tail[0] = (float)tf;      // inhibit_time
    out_tail[1] = (float)winner;  // winner
}

// ---------------------------------------------------------------------------
// Kernel 5: expand bit-planes -> bank_spk floats, gated by t <= t_f.
// Fully coalesced elementwise over T*B*N; plane reads are uniform within each 16-lane half.
// ---------------------------------------------------------------------------
__global__ __launch_bounds__(256)
void k_expand(const unsigned short* __restrict__ maskp,
              const int* __restrict__ scal,
              float* __restrict__ out)
{
    const size_t idx = (size_t)blockIdx.x * 256 + threadIdx.x;
    const int tf = scal[0];
    const int n  = (int)(idx & 63);
    const size_t tb = idx >> 6;          // t*B + b
    const int t  = (int)(tb >> 10);      // B = 1024
    const bool keep = (tf < 0) || (t <= tf);
    float vout = 0.0f;
    if (keep) {
        const unsigned piece = maskp[(size_t)(n >> 4) * TB + tb];
        vout = (float)((piece >> (n & 15)) & 1u);
    }
    out[idx] = vout;
}

// ---------------------------------------------------------------------------
// Kernel 6: final readout spikes. Input zeroed for t > t_f (post-inhibit
// decay), matching the reference trajectory exactly.
// ---------------------------------------------------------------------------
__global__ __launch_bounds__(256)
void k_rospk(const unsigned short* __restrict__ maskp,
             const int* __restrict__ scal,
             float* __restrict__ out)
{
    const int b = blockIdx.x * 256 + threadIdx.x;
    const int tf = scal[0];
    float m = 0.0f;
    for (int t = 0; t < T_STEPS; ++t) {
        const bool keep = (tf < 0) || (t <= tf);
        float cur = 0.0f;
        if (keep)
            cur = (float)__popcll(load_mask(maskp, (size_t)t * B_SIZE + b));
        const float reset = (m > THRESH) ? THRESH : 0.0f;
        m = BETA * m + cur - reset;
        out[(size_t)t * B_SIZE + b] = (m > THRESH) ? 1.0f : 0.0f;
    }
}

// ---------------------------------------------------------------------------
extern "C" void kernel_launch(void* const* d_in, const int* in_sizes, int n_in,
                              void* d_out, int out_size, void* d_ws, size_t ws_size,
                              hipStream_t stream)
{
    const float* spike  = (const float*)d_in[0];   // [T,B,2]
    const float* W      = (const float*)d_in[1];   // [N,2]
    const int*   delays = (const int*)d_in[2];     // [N]

    // Workspace layout (~8.02 MB total)
    char* ws = (char*)d_ws;
    unsigned short*     maskp  = (unsigned short*)ws;                   // 4 planes of T*B u16
    unsigned long long* active = (unsigned long long*)(ws + 4 * TB * 2);
    unsigned*           fire   = (unsigned*)((char*)active + (size_t)T_STEPS * 8);
    int*                scal   = (int*)((char*)fire + (size_t)T_STEPS * 4);

    float* out_bank = (float*)d_out;                                   // [T,B,N]
    float* out_ro   = out_bank + TB * N_NEUR;                          // [T,B]
    float* out_tail = out_ro   + TB;                                   // [2]

    // 1) bank LIF via WMMA -> spike bit-planes
    k_bank<<<B_SIZE / 16, 128, 0, stream>>>(spike, W, maskp);
    // 2) per-t active mask (OR over b) + zero fire flags
    k_active<<<T_STEPS, 256, 0, stream>>>(maskp, active, fire);
    // 3) raw readout sim -> fire flags
    k_ro_fire<<<1, 1024, 0, stream>>>(maskp, fire);
    // 4) t_f + winner scalars
    k_scalars<<<1, 32, 0, stream>>>(active, fire, delays, scal, out_tail);
    // 5) expand bit-planes to bank_spk floats (gated by t <= t_f)
    {
        const size_t total = TB * N_NEUR;
        k_expand<<<(unsigned)(total / 256), 256, 0, stream>>>(maskp, scal, out_bank);
    }
    // 6) final readout spikes
    k_rospk<<<B_SIZE / 256, 256, 0, stream>>>(maskp, scal, out_ro);
}